// DepthManipulator_62277025792073
// MI455X (gfx1250) — compile-verified
//
#include <hip/hip_runtime.h>
#include <hip/hip_bf16.h>

typedef float v2f __attribute__((ext_vector_type(2)));
typedef float v8f __attribute__((ext_vector_type(8)));

#define W 256
#define NPIX (W * W)

// radix sort config: 4-bit digits, 8 passes
#define RD 16          // bins
#define RSTR 257       // LDS row stride (odd mod 64 -> conflict-free serial scans)
#define RTPB 256       // threads per block
#define RIPT 16        // items per thread
#define REPB (RTPB * RIPT)   // 4096 elements per block
#define RBPB (NPIX / REPB)   // 16 blocks per batch

__device__ __forceinline__ unsigned zkey(float f) {
    unsigned u = __float_as_uint(f);
    return (u & 0x80000000u) ? ~u : (u | 0x80000000u);
}

// ---- 1) M = K @ RT_cam2 @ RTinv_cam1 @ K_inv, per batch ----
__global__ void k_matchain(const float* __restrict__ Km, const float* __restrict__ Kinv,
                           const float* __restrict__ RTinv, const float* __restrict__ RT2,
                           float* __restrict__ M) {
    int b = blockIdx.x, t = threadIdx.x;   // 16 threads
    int r = t >> 2, c = t & 3;
    __shared__ float T1[16], T2[16];
    const float* rt2 = RT2 + b * 16;
    const float* rti = RTinv + b * 16;
    const float* k   = Km + b * 16;
    const float* ki  = Kinv + b * 16;
    float s = 0.f;
    for (int q = 0; q < 4; ++q) s += rt2[r * 4 + q] * rti[q * 4 + c];
    T1[t] = s; __syncthreads();
    s = 0.f;
    for (int q = 0; q < 4; ++q) s += k[r * 4 + q] * T1[q * 4 + c];
    T2[t] = s; __syncthreads();
    s = 0.f;
    for (int q = 0; q < 4; ++q) s += T2[r * 4 + q] * ki[q * 4 + c];
    M[b * 16 + t] = s;
}

// ---- 2) per-pixel projection via V_WMMA_F32_16X16X4_F32 ----
// A (16x4, M padded): lanes 0-15 hold {K0,K1}, lanes 16-31 hold {K2,K3}, lane&15 = row.
// B (4x16, pixel vecs): lane&15 = column (pixel); lanes 0-15 supply K0,K1; lanes 16-31 K2,K3.
// D (16x16): lane n<16, VGPR r -> D[r][n]; rows 0..2 = x,y,z of pixel n.
__global__ void k_transform(const float* __restrict__ depth, const float* __restrict__ Mg,
                            unsigned* __restrict__ keys, unsigned* __restrict__ vals,
                            unsigned* __restrict__ packed, float* __restrict__ zout) {
    int b = blockIdx.y;
    int lane = threadIdx.x & 31;
    int wave = threadIdx.x >> 5;
    int row = lane & 15;

    v2f a; a.x = 0.0f; a.y = 0.0f;
    if (row < 4) {
        int cb = (lane < 16) ? 0 : 2;
        a.x = Mg[b * 16 + row * 4 + cb];
        a.y = Mg[b * 16 + row * 4 + cb + 1];
    }

    const float* dp = depth + (size_t)b * NPIX;
    size_t bo = (size_t)b * NPIX;
    int wbase = (blockIdx.x * 8 + wave) * 128;   // 8 waves/block, 128 px/wave

    for (int i = 0; i < 8; ++i) {
        int p = wbase + i * 16 + row;
        float d = dp[p];
        if (i < 7) __builtin_prefetch(dp + p + 16, 0, 1);   // global_prefetch_b8
        int rr = p >> 8, cc = p & 255;
        float X = -1.0f + (float)cc * (2.0f / 255.0f);
        float Y =  1.0f - (float)rr * (2.0f / 255.0f);
        v2f bv;
        bv.x = (lane < 16) ? X * d : -d;
        bv.y = (lane < 16) ? Y * d : 1.0f;
        v8f acc = {};
        v8f dm = __builtin_amdgcn_wmma_f32_16x16x4_f32(false, a, false, bv,
                                                       (short)0, acc, false, false);
        if (lane < 16) {
            float x = dm[0], y = dm[1], z = dm[2];
            float zn = -z;
            zout[bo + p] = zn;
            keys[bo + p] = zkey(zn);
            vals[bo + p] = (unsigned)p;
            bool me = fabsf(z) < 0.01f;
            float s0 = me ? -10.0f : x / zn;
            float s1 = -(me ? -10.0f : y / zn);
            float t0 = (s0 + 1.0f) * 128.0f;
            float t1 = (s1 + 1.0f) * 128.0f;
            unsigned mb = (t0 < 0.0f || t0 > 255.0f || t1 < 0.0f || t1 > 255.0f) ? 1u : 0u;
            int xi = (int)t0; xi = xi < 0 ? 0 : (xi > 255 ? 255 : xi);
            int yi = (int)t1; yi = yi < 0 ? 0 : (yi > 255 ? 255 : yi);
            packed[bo + p] = (unsigned)(yi * W + xi) | (mb << 16);
        }
    }
}

// ---- 3) stable LSD radix sort (key, val) per batch ----
__global__ void k_radix_hist(const unsigned* __restrict__ keysIn, unsigned* __restrict__ hist,
                             int shift) {
    __shared__ unsigned cnt[RD * RSTR];
    int b = blockIdx.y, blk = blockIdx.x, t = threadIdx.x;
    for (int j = t; j < RD * RSTR; j += RTPB) cnt[j] = 0;
    __syncthreads();
    const unsigned* kp = keysIn + (size_t)b * NPIX + blk * REPB + t * RIPT;
#pragma unroll
    for (int i = 0; i < RIPT; ++i) {
        unsigned d = (kp[i] >> shift) & (RD - 1);
        cnt[d * RSTR + t]++;
    }
    __syncthreads();
    if (t < RD) {
        unsigned s = 0;
        for (int j = 0; j < RTPB; ++j) s += cnt[t * RSTR + j];
        hist[(b * RBPB + blk) * RD + t] = s;
    }
}

// per batch: exclusive scan over 256 entries in (bin-major, block-minor) order
__global__ void k_scan_offsets(const unsigned* __restrict__ hist, unsigned* __restrict__ bboff) {
    int b = blockIdx.x, t = threadIdx.x;
    int bin = t >> 4, blk = t & 15;
    unsigned v = hist[(b * RBPB + blk) * RD + bin];
    __shared__ unsigned s[256];
    s[t] = v;
    __syncthreads();
    for (int off = 1; off < 256; off <<= 1) {
        unsigned x = (t >= off) ? s[t - off] : 0u;
        __syncthreads();
        s[t] += x;
        __syncthreads();
    }
    bboff[(b * RBPB + blk) * RD + bin] = s[t] - v;
}

__global__ void k_radix_scatter(const unsigned* __restrict__ keysIn, const unsigned* __restrict__ valsIn,
                                unsigned* __restrict__ keysOut, unsigned* __restrict__ valsOut,
                                const unsigned* __restrict__ bboff, int shift) {
    __shared__ unsigned cnt[RD * RSTR];
    int b = blockIdx.y, blk = blockIdx.x, t = threadIdx.x;
    for (int j = t; j < RD * RSTR; j += RTPB) cnt[j] = 0;
    __syncthreads();
    size_t base = (size_t)b * NPIX + blk * REPB + t * RIPT;
    const unsigned* kp = keysIn + base;
#pragma unroll
    for (int i = 0; i < RIPT; ++i) {
        unsigned d = (kp[i] >> shift) & (RD - 1);
        cnt[d * RSTR + t]++;
    }
    __syncthreads();
    if (t < RD) {   // convert to absolute destinations: bboff + exclusive scan over threads
        unsigned run = bboff[(b * RBPB + blk) * RD + t];
        for (int j = 0; j < RTPB; ++j) {
            unsigned v = cnt[t * RSTR + j];
            cnt[t * RSTR + j] = run;
            run += v;
        }
    }
    __syncthreads();
    const unsigned* vp = valsIn + base;
    unsigned* ko = keysOut + (size_t)b * NPIX;
    unsigned* vo = valsOut + (size_t)b * NPIX;
#pragma unroll
    for (int i = 0; i < RIPT; ++i) {
        unsigned k = kp[i], v = vp[i];
        unsigned d = (k >> shift) & (RD - 1);
        unsigned pos = cnt[d * RSTR + t]++;
        ko[pos] = k;
        vo[pos] = v;
    }
}

// ---- 4) winner init + scatter-max of rank ----
__global__ void k_winner_init(int* __restrict__ winner, int total) {
    int tid = blockIdx.x * blockDim.x + threadIdx.x;
    if (tid < total) winner[tid] = -1;
}

__global__ void k_scatter_max(const unsigned* __restrict__ valsSorted,
                              const unsigned* __restrict__ packed,
                              int* __restrict__ winner, int total) {
    int tid = blockIdx.x * blockDim.x + threadIdx.x;
    if (tid >= total) return;
    int b = tid >> 16;              // NPIX == 65536
    int j = tid & (NPIX - 1);       // rank within batch
    unsigned p = valsSorted[tid];   // pixel at rank j
    unsigned f = packed[(size_t)b * NPIX + p] & 0xFFFFu;
    atomicMax(winner + (size_t)b * NPIX + f, j);
}

// ---- 5) final gather ----
__global__ void k_gather(const int* __restrict__ winner, const unsigned* __restrict__ valsSorted,
                         const unsigned* __restrict__ packed, float* __restrict__ out, int total) {
    int tid = blockIdx.x * blockDim.x + threadIdx.x;
    if (tid >= total) return;
    int b = tid >> 16;
    int t = tid & (NPIX - 1);
    int j = winner[tid];
    float ch0 = -2.0f, ch1 = -2.0f;
    if (j >= 0) {
        unsigned p = valsSorted[(size_t)b * NPIX + j];
        // reference quirk: maskf indexed by rank j (as unsorted position)
        float mf = ((packed[(size_t)b * NPIX + j] >> 16) & 1u) ? 4.0f : 0.0f;
        float g0 = -1.0f + (float)(p & 255u) * (2.0f / 255.0f);
        float g1 =  1.0f - (float)(p >> 8)   * (2.0f / 255.0f);
        ch0 = g0 + mf;
        ch1 = -g1 + mf;
    }
    out[(size_t)b * 2 * NPIX + t] = ch0;
    out[(size_t)b * 2 * NPIX + NPIX + t] = ch1;
}

extern "C" void kernel_launch(void* const* d_in, const int* in_sizes, int n_in,
                              void* d_out, int out_size, void* d_ws, size_t ws_size,
                              hipStream_t stream) {
    (void)n_in; (void)out_size; (void)ws_size;
    const float* depth = (const float*)d_in[0];
    const float* Km    = (const float*)d_in[1];
    const float* Kinv  = (const float*)d_in[2];
    const float* RTinv = (const float*)d_in[3];
    const float* RT2   = (const float*)d_in[4];
    int bs = in_sizes[0] / NPIX;
    size_t E = (size_t)bs * NPIX;

    unsigned* keysA  = (unsigned*)d_ws;
    unsigned* valsA  = keysA + E;
    unsigned* keysB  = valsA + E;
    unsigned* valsB  = keysB + E;
    unsigned* packed = valsB + E;
    int*      winner = (int*)(packed + E);
    unsigned* hist   = (unsigned*)(winner + E);
    unsigned* bboff  = hist + (size_t)bs * RBPB * RD;
    float*    M      = (float*)(bboff + (size_t)bs * RBPB * RD);

    float* out  = (float*)d_out;
    float* zout = out + (size_t)bs * 2 * NPIX;

    k_matchain<<<bs, 16, 0, stream>>>(Km, Kinv, RTinv, RT2, M);
    k_transform<<<dim3(NPIX / (8 * 128), bs), 256, 0, stream>>>(depth, M, keysA, valsA,
                                                                packed, zout);
    for (int pass = 0; pass < 8; ++pass) {
        int shift = pass * 4;
        const unsigned* ki = (pass & 1) ? keysB : keysA;
        const unsigned* vi = (pass & 1) ? valsB : valsA;
        unsigned* ko = (pass & 1) ? keysA : keysB;
        unsigned* vo = (pass & 1) ? valsA : valsB;
        k_radix_hist<<<dim3(RBPB, bs), RTPB, 0, stream>>>(ki, hist, shift);
        k_scan_offsets<<<bs, 256, 0, stream>>>(hist, bboff);
        k_radix_scatter<<<dim3(RBPB, bs), RTPB, 0, stream>>>(ki, vi, ko, vo, bboff, shift);
    }
    // after 8 passes sorted (key,val) are back in keysA/valsA
    int total = (int)E;
    int thr = 256, blks = (total + thr - 1) / thr;
    k_winner_init<<<blks, thr, 0, stream>>>(winner, total);
    k_scatter_max<<<blks, thr, 0, stream>>>(valsA, packed, winner, total);
    k_gather<<<blks, thr, 0, stream>>>(winner, valsA, packed, out, total);
}